// WSA_53180285059591
// MI455X (gfx1250) — compile-verified
//
#include <hip/hip_runtime.h>
#include <hip/hip_bf16.h>

// MI455X / gfx1250, wave32. Fused WSA block:
//   t   = BN(x @ w_in^T + b_in)              (pointwise, WMMA f32)
//   per 8x8 window: q = g@w_mask^T+b ; s = softmax(q q^T) ; y = s v   (WMMA f32)
//   out = y @ w_out^T + b_out                (pointwise, WMMA f32)
// All GEMMs use V_WMMA_F32_16X16X4_F32 (exact fp32; problem is HBM-bound:
// ~10.7 GFLOP vs ~128MB minimum traffic @ 23.3 TB/s). x loads and final out
// stores are non-temporal so the re-read intermediates t & y stay in the
// 192MB L2.

typedef float v2f __attribute__((ext_vector_type(2)));
typedef float v4f __attribute__((ext_vector_type(4)));
typedef float v8f __attribute__((ext_vector_type(8)));

#define LDA 65   // padded row stride for 64x64 LDS matrices (bank-conflict avoidance)

// ---- wave-level 64x64x64 GEMM, all operands in LDS (row stride LDA) ----
// C[m][n] = sum_k A[m][k] * (TB ? B[n][k] : B[k][n])  (+ bias[n])
// f32 WMMA 16x16x4 fragment layout:
//   A: lane<16 -> M=lane, K=kb{+0,+1} ; lane>=16 -> M=lane-16, K=kb{+2,+3}
//   B: lane<16 -> N=lane, same K split ; C/D: vgpr j -> M=j(+8 for hi half), N=lane&15
template <bool TB>
__device__ __forceinline__ void wave_gemm64(const float* __restrict__ A,
                                            const float* __restrict__ B,
                                            float* __restrict__ C,
                                            const float* __restrict__ bias) {
  const int lane = threadIdx.x & 31;
  const int half = lane >> 4;   // 0: K+{0,1}, 1: K+{2,3}
  const int ln   = lane & 15;
  for (int mt = 0; mt < 4; ++mt) {
    v8f acc[4] = {};
    const float* Arow = A + (mt * 16 + ln) * LDA;
    for (int kt = 0; kt < 16; ++kt) {
      const int kb = kt * 4 + half * 2;
      v2f a;
      a.x = Arow[kb];
      a.y = Arow[kb + 1];
#pragma unroll
      for (int nt = 0; nt < 4; ++nt) {
        const int n = nt * 16 + ln;
        v2f bv;
        if (TB) { bv.x = B[n * LDA + kb];     bv.y = B[n * LDA + kb + 1]; }
        else    { bv.x = B[kb * LDA + n];     bv.y = B[(kb + 1) * LDA + n]; }
        acc[nt] = __builtin_amdgcn_wmma_f32_16x16x4_f32(
            false, a, false, bv, (short)0, acc[nt], false, false);
      }
    }
#pragma unroll
    for (int nt = 0; nt < 4; ++nt) {
#pragma unroll
      for (int j = 0; j < 8; ++j) {
        const int m = mt * 16 + j + half * 8;
        const int n = nt * 16 + ln;
        float v = acc[nt][j];
        if (bias) v += bias[n];
        C[m * LDA + n] = v;
      }
    }
  }
}

// ---------------- pointwise 1x1 conv (+optional folded BN) ----------------
// out[b][o][p] = sum_c in[b][c][p] * (w[o][c]*scale[o]) + biasEff[o]
// 256 threads = 8 waves, 128 pixels per block; each wave: 16-pixel M tile.
// Input loads are always non-temporal (x is streamed once; y is last-use).
// Output stores are non-temporal only for the final conv (never re-read).
__global__ void __launch_bounds__(256)
wsa_pointwise(const float* __restrict__ in, const float* __restrict__ w,
              const float* __restrict__ bias, const float* __restrict__ bn_g,
              const float* __restrict__ bn_b, const float* __restrict__ bn_m,
              const float* __restrict__ bn_v, float* __restrict__ out,
              int use_bn, int nt_store) {
  __shared__ __align__(16) float sbuf[8448];      // inT[128][65] then outT[64][132]
  __shared__ __align__(16) float Wt[64 * LDA];    // Wt[k][n] = w[n][k]*scale[n]
  __shared__ float sscale[64];
  __shared__ float sbias[64];

  const int tid  = threadIdx.x;
  const int p0   = blockIdx.x * 128;          // global pixel base (b,h,w flattened)
  const int bch  = p0 >> 16;                  // batch (h*w = 65536)
  const int hw0  = p0 & 65535;
  const float* inG  = in  + (size_t)bch * 64 * 65536;
  float*       outG = out + (size_t)bch * 64 * 65536;

  if (tid < 64) {
    float sc = 1.0f, sb;
    if (use_bn) {
      const float inv = bn_g[tid] * rsqrtf(bn_v[tid] + 1e-5f);
      sc = inv;
      sb = (bias[tid] - bn_m[tid]) * inv + bn_b[tid];
    } else {
      sb = bias[tid];
    }
    sscale[tid] = sc;
    sbias[tid]  = sb;
  }
  __syncthreads();

  // stage weights (transposed + scaled), v4f reads along k
  for (int i = tid; i < 1024; i += 256) {
    const int n = i >> 4, k4 = (i & 15) * 4;
    const v4f vv = *(const v4f*)&w[n * 64 + k4];
    const float sc = sscale[n];
    Wt[(k4 + 0) * LDA + n] = vv.x * sc;
    Wt[(k4 + 1) * LDA + n] = vv.y * sc;
    Wt[(k4 + 2) * LDA + n] = vv.z * sc;
    Wt[(k4 + 3) * LDA + n] = vv.w * sc;
  }
  // stage input tile: coalesced NT v4f reads, write token-major inT[pix][ch]
  for (int i = tid; i < 2048; i += 256) {
    const int c = i >> 5, p4 = i & 31;
    const v4f vv =
        __builtin_nontemporal_load((const v4f*)&inG[c * 65536 + hw0 + p4 * 4]);
    const int pb = p4 * 4;
    sbuf[(pb + 0) * LDA + c] = vv.x;
    sbuf[(pb + 1) * LDA + c] = vv.y;
    sbuf[(pb + 2) * LDA + c] = vv.z;
    sbuf[(pb + 3) * LDA + c] = vv.w;
  }
  __syncthreads();

  // per-wave 16x64x64 GEMM
  const int wave = tid >> 5, lane = tid & 31;
  const int half = lane >> 4, ln = lane & 15;
  const float* Arow = &sbuf[(wave * 16 + ln) * LDA];
  v8f acc[4] = {};
  for (int kt = 0; kt < 16; ++kt) {
    const int kb = kt * 4 + half * 2;
    v2f a;
    a.x = Arow[kb];
    a.y = Arow[kb + 1];
#pragma unroll
    for (int nt = 0; nt < 4; ++nt) {
      const int n = nt * 16 + ln;
      v2f bv;
      bv.x = Wt[kb * LDA + n];
      bv.y = Wt[(kb + 1) * LDA + n];
      acc[nt] = __builtin_amdgcn_wmma_f32_16x16x4_f32(
          false, a, false, bv, (short)0, acc[nt], false, false);
    }
  }
  __syncthreads();   // done reading inT; sbuf becomes outT[64][132]

#pragma unroll
  for (int nt = 0; nt < 4; ++nt) {
#pragma unroll
    for (int j = 0; j < 8; ++j) {
      const int m = wave * 16 + j + half * 8;     // pixel within 128
      const int n = nt * 16 + ln;                 // out channel
      sbuf[n * 132 + m] = acc[nt][j] + sbias[n];
    }
  }
  __syncthreads();

  // coalesced v4f stores, channel-major
  for (int i = tid; i < 2048; i += 256) {
    const int c = i >> 5, p4 = i & 31;
    const v4f vv = *(const v4f*)&sbuf[c * 132 + p4 * 4];
    float* dst = &outG[c * 65536 + hw0 + p4 * 4];
    if (nt_store) __builtin_nontemporal_store(vv, (v4f*)dst);
    else          *(v4f*)dst = vv;
  }
}

// ---------------- per-window attention (1 wave / window) ----------------
__global__ void __launch_bounds__(32)
wsa_attn(const float* __restrict__ t, const float* __restrict__ wm,
         const float* __restrict__ bm, float* __restrict__ y) {
  __shared__ __align__(16) float bufA[64 * LDA];  // g  -> s (scores)
  __shared__ __align__(16) float bufQ[64 * LDA];  // q  -> y
  __shared__ __align__(16) float bufB[64 * LDA];  // Wm^T -> v
  __shared__ float sbias[64];

  const int lane = threadIdx.x;
  const int win  = blockIdx.x;
  const int b    = win >> 10;
  const int rem  = win & 1023;
  const int hi   = rem >> 5;     // window row index  (h / 8)
  const int wj   = rem & 31;     // window col index  (w / 8)
  const float* tb = t + (size_t)b * 64 * 65536;

  // stage w_mask^T (v4f reads along k) and bias
  for (int i = lane; i < 1024; i += 32) {
    const int n = i >> 4, k4 = (i & 15) * 4;
    const v4f vv = *(const v4f*)&wm[n * 64 + k4];
    bufB[(k4 + 0) * LDA + n] = vv.x;
    bufB[(k4 + 1) * LDA + n] = vv.y;
    bufB[(k4 + 2) * LDA + n] = vv.z;
    bufB[(k4 + 3) * LDA + n] = vv.w;
  }
  for (int i = lane; i < 64; i += 32) sbias[i] = bm[i];

  // stage g: token = ws_col*8 + ws_row ; rows/cols sampled every 2 with wrap.
  // One aligned v4f covers a pair of sampled columns (use .x/.z). The pair
  // base is a multiple of 4 columns, so the >=256 wrap is uniform per pair.
  for (int i = lane; i < 2048; i += 32) {
    const int ch = i >> 5;
    const int s5 = i & 31;
    const int up = s5 & 3;       // column pair index (ws_col = 2*up, 2*up+1)
    const int r  = s5 >> 2;      // ws_row
    int gr = hi * 8 + 2 * r; if (gr >= 256) gr -= 8;
    int gc = wj * 8 + 4 * up; if (gc >= 256) gc -= 8;
    const int u0 = 2 * up;
    const v4f vv = *(const v4f*)&tb[(ch * 256 + gr) * 256 + gc];
    bufA[(u0 * 8 + r) * LDA + ch]       = vv.x;
    bufA[((u0 + 1) * 8 + r) * LDA + ch] = vv.z;
  }

  // q = g @ w_mask^T + b_mask
  wave_gemm64<false>(bufA, bufB, bufQ, sbias);

  // stage v (overwrites Wm^T; LDS ops of one wave are in-order):
  // token = ws_row*8 + ws_col, contiguous rows -> aligned v4f loads
  for (int i = lane; i < 1024; i += 32) {
    const int ch  = i >> 4;
    const int seg = i & 15;
    const int r   = seg >> 1;
    const int cs  = (seg & 1) * 4;
    const v4f vv = *(const v4f*)&tb[(ch * 256 + hi * 8 + r) * 256 + wj * 8 + cs];
    const int tokb = r * 8 + cs;
    bufB[(tokb + 0) * LDA + ch] = vv.x;
    bufB[(tokb + 1) * LDA + ch] = vv.y;
    bufB[(tokb + 2) * LDA + ch] = vv.z;
    bufB[(tokb + 3) * LDA + ch] = vv.w;
  }

  // s = q @ q^T (into bufA, g no longer needed)
  wave_gemm64<true>(bufQ, bufQ, bufA, nullptr);

  // row softmax: 2 rows per lane
  for (int rr = lane; rr < 64; rr += 32) {
    float mx = -3.0e38f;
    for (int j = 0; j < 64; ++j) mx = fmaxf(mx, bufA[rr * LDA + j]);
    float sum = 0.0f;
    for (int j = 0; j < 64; ++j) {
      const float e = __expf(bufA[rr * LDA + j] - mx);
      bufA[rr * LDA + j] = e;
      sum += e;
    }
    const float rs = 1.0f / sum;
    for (int j = 0; j < 64; ++j) bufA[rr * LDA + j] *= rs;
  }

  // y = s @ v (into bufQ, q no longer needed)
  wave_gemm64<false>(bufA, bufB, bufQ, nullptr);

  // store back: token i=(a,b2) -> (h = hi*8+a, w = wj*8+b2)  [reference's
  // final transpose maps q-token dims (ws_col, ws_row) onto (h, w)]
  // aligned v4f stores along w
  float* yb = y + (size_t)b * 64 * 65536;
  for (int i = lane; i < 1024; i += 32) {
    const int ch  = i >> 4;
    const int seg = i & 15;
    const int r   = seg >> 1;      // a  (h offset)
    const int cs  = (seg & 1) * 4; // b2 (w offset base)
    const int tokb = r * 8 + cs;
    v4f vv;
    vv.x = bufQ[(tokb + 0) * LDA + ch];
    vv.y = bufQ[(tokb + 1) * LDA + ch];
    vv.z = bufQ[(tokb + 2) * LDA + ch];
    vv.w = bufQ[(tokb + 3) * LDA + ch];
    *(v4f*)&yb[(ch * 256 + hi * 8 + r) * 256 + wj * 8 + cs] = vv;
  }
}

extern "C" void kernel_launch(void* const* d_in, const int* in_sizes, int n_in,
                              void* d_out, int out_size, void* d_ws, size_t ws_size,
                              hipStream_t stream) {
  const float* x      = (const float*)d_in[0];
  const float* w_in   = (const float*)d_in[1];
  const float* b_in   = (const float*)d_in[2];
  const float* bn_g   = (const float*)d_in[3];
  const float* bn_b   = (const float*)d_in[4];
  const float* bn_m   = (const float*)d_in[5];
  const float* bn_v   = (const float*)d_in[6];
  const float* w_mask = (const float*)d_in[7];
  const float* b_mask = (const float*)d_in[8];
  const float* w_out  = (const float*)d_in[9];
  const float* b_out  = (const float*)d_in[10];
  float* out = (float*)d_out;

  // workspace: t (4*64*256*256 f32 = 64MB) + y (64MB)
  float* t  = (float*)d_ws;
  float* yb = t + (size_t)4 * 64 * 256 * 256;

  // 262144 pixels / 128 per block
  wsa_pointwise<<<2048, 256, 0, stream>>>(x, w_in, b_in, bn_g, bn_b, bn_m, bn_v,
                                          t, 1, 0);
  // 4 * 32 * 32 windows, one wave each
  wsa_attn<<<4096, 32, 0, stream>>>(t, w_mask, b_mask, yb);
  wsa_pointwise<<<2048, 256, 0, stream>>>(yb, w_out, b_out, nullptr, nullptr, nullptr,
                                          nullptr, out, 0, 1);
}